// GIN0_58248346468611
// MI455X (gfx1250) — compile-verified
//
#include <hip/hip_runtime.h>
#include <hip/hip_bf16.h>

#define NN 50000
#define EE 800000
#define HH 128
#define LL 5
#define GG 64
#define NH (NN * HH)

typedef __attribute__((ext_vector_type(2))) float v2f;
typedef __attribute__((ext_vector_type(8))) float v8f;

// ---------------------------------------------------------------------------
// Edge aggregation: agg[dst] += x[src].  One wave (32 lanes) per edge; each
// lane owns 4 contiguous channels -> float4 coalesced gather + 4 f32 atomics.
// ---------------------------------------------------------------------------
__global__ __launch_bounds__(256) void gin_scatter_add(
    const float* __restrict__ x, const long long* __restrict__ ei,
    float* __restrict__ agg, int nedges) {
  long long t = (long long)blockIdx.x * blockDim.x + threadIdx.x;
  int edge = (int)(t >> 5);
  if (edge >= nedges) return;
  int lane = (int)(t & 31);
  int s = (int)ei[edge];                       // edge_index[0][e] = src
  int d = (int)ei[(long long)nedges + edge];   // edge_index[1][e] = dst
  const float4* xs = (const float4*)(x + (long long)s * HH);
  float4 v = xs[lane];
  float* ap = agg + (long long)d * HH + lane * 4;
  atomicAdd(ap + 0, v.x);
  atomicAdd(ap + 1, v.y);
  atomicAdd(ap + 2, v.z);
  atomicAdd(ap + 3, v.w);
}

// agg = (1+eps[layer]) * xin + agg   (elementwise fuse of GIN self term)
__global__ __launch_bounds__(256) void gin_combine(
    const float* __restrict__ xin, const float* __restrict__ epsArr, int layer,
    float* __restrict__ agg, int n) {
  int i = blockIdx.x * blockDim.x + threadIdx.x;
  if (i < n) agg[i] = fmaf(1.0f + epsArr[layer], xin[i], agg[i]);
}

// ---------------------------------------------------------------------------
// Fp32 WMMA GEMM: out[M,128] = act(A[M,128] @ W[128,128] + bias)
// Block = 256 threads = 8 waves; block owns 16 rows, wave w owns cols 16w..16w+15.
// Wave computes a 16x16 tile with 32 x V_WMMA_F32_16X16X4_F32 along K.
// A layout (16x4 f32): lanes 0-15 = M rows; VGPR0/1 = K pair (2*lanehalf)+0/1.
// C/D layout: VGPR r -> M = r + 8*lanehalf, N = lane&15.
// act: 0 = none, 1 = relu, 2 = tanh
// ---------------------------------------------------------------------------
__global__ __launch_bounds__(256) void gin_gemm128_wmma(
    const float* __restrict__ A, const float* __restrict__ W,
    const float* __restrict__ bias, float* __restrict__ out,
    int nrows, int act) {
  const int wave  = threadIdx.x >> 5;
  const int lane  = threadIdx.x & 31;
  const int lhalf = lane >> 4;   // 0: K pair {0,1} / M 0-7 ; 1: K pair {2,3} / M 8-15
  const int l16   = lane & 15;
  const int m0 = blockIdx.x * 16;
  const int n0 = wave * 16;

  int row = m0 + l16;
  int rc  = row < nrows ? row : nrows - 1;     // clamp (no EXEC divergence)
  const float* Ap = A + (long long)rc * HH;

  v8f acc = {0.f, 0.f, 0.f, 0.f, 0.f, 0.f, 0.f, 0.f};
#pragma unroll
  for (int k = 0; k < HH; k += 4) {
    int ka = k + 2 * lhalf;
    v2f a;
    a.x = Ap[ka + 0];
    a.y = Ap[ka + 1];
    v2f b;
    b.x = W[(ka + 0) * HH + n0 + l16];
    b.y = W[(ka + 1) * HH + n0 + l16];
    acc = __builtin_amdgcn_wmma_f32_16x16x4_f32(
        /*neg_a=*/false, a, /*neg_b=*/false, b,
        /*c_mod=*/(short)0, acc, /*reuse_a=*/false, /*reuse_b=*/false);
  }

  const int ocol = n0 + l16;
  const float bv = bias[ocol];
#pragma unroll
  for (int r = 0; r < 8; ++r) {
    int orow = m0 + r + 8 * lhalf;
    if (orow < nrows) {
      float v = acc[r] + bv;
      if (act == 1) v = fmaxf(v, 0.f);
      else if (act == 2) v = tanhf(v);
      out[(long long)orow * HH + ocol] = v;
    }
  }
}

// ---------------------------------------------------------------------------
// BatchNorm statistics: stats[0:128] += sum, stats[128:256] += sumsq.
// blockDim = 128 (one thread per channel, coalesced row reads), one atomic
// pair per block per channel.
// ---------------------------------------------------------------------------
__global__ __launch_bounds__(128) void gin_bn_stats(
    const float* __restrict__ h, float* __restrict__ stats, int nrows) {
  int c = threadIdx.x;
  int rows_per_block = (nrows + gridDim.x - 1) / gridDim.x;
  int r0 = blockIdx.x * rows_per_block;
  int r1 = min(r0 + rows_per_block, nrows);
  float s = 0.f, s2 = 0.f;
  for (int r = r0; r < r1; ++r) {
    float v = h[(long long)r * HH + c];
    s += v;
    s2 += v * v;
  }
  atomicAdd(&stats[c], s);
  atomicAdd(&stats[HH + c], s2);
}

// h = bn(t)  (layer 0)   or   h = h + relu(bn(t))  (residual layers)
__global__ __launch_bounds__(256) void gin_bn_apply(
    const float* __restrict__ t, const float* __restrict__ stats,
    const float* __restrict__ gamma, const float* __restrict__ beta,
    float* __restrict__ h, int nrows, int residual) {
  int i = blockIdx.x * blockDim.x + threadIdx.x;
  if (i >= nrows * HH) return;
  int c = i & (HH - 1);
  float invN = 1.0f / (float)nrows;
  float mu = stats[c] * invN;
  float var = stats[HH + c] * invN - mu * mu;
  float inv = rsqrtf(var + 1e-5f);
  float v = (t[i] - mu) * inv * gamma[c] + beta[c];
  if (residual) h[i] = h[i] + fmaxf(v, 0.f);
  else h[i] = v;
}

// ---------------------------------------------------------------------------
// Mean pool by graph id
// ---------------------------------------------------------------------------
__global__ __launch_bounds__(256) void gin_pool_accum(
    const float* __restrict__ y, const long long* __restrict__ batch,
    float* __restrict__ sums, int nrows) {
  int i = blockIdx.x * blockDim.x + threadIdx.x;
  if (i >= nrows * HH) return;
  int r = i >> 7, c = i & (HH - 1);
  int g = (int)batch[r];
  atomicAdd(&sums[g * HH + c], y[i]);
}

__global__ __launch_bounds__(256) void gin_pool_count(
    const long long* __restrict__ batch, float* __restrict__ counts, int nrows) {
  int i = blockIdx.x * blockDim.x + threadIdx.x;
  if (i < nrows) atomicAdd(&counts[(int)batch[i]], 1.0f);
}

__global__ __launch_bounds__(256) void gin_pool_final(
    const float* __restrict__ sums, const float* __restrict__ counts,
    float* __restrict__ out) {
  int i = blockIdx.x * blockDim.x + threadIdx.x;
  if (i < GG * HH) {
    int g = i >> 7;
    out[i] = sums[i] / fmaxf(counts[g], 1.0f);
  }
}

// ---------------------------------------------------------------------------
extern "C" void kernel_launch(void* const* d_in, const int* in_sizes, int n_in,
                              void* d_out, int out_size, void* d_ws, size_t ws_size,
                              hipStream_t stream) {
  const float*     x     = (const float*)d_in[0];
  const long long* ei    = (const long long*)d_in[1];   // int64 [2,E]
  const long long* batch = (const long long*)d_in[2];   // int64 [N]
  const float*     W1    = (const float*)d_in[3];       // [L,H,H]
  const float*     b1    = (const float*)d_in[4];       // [L,H]
  const float*     W2    = (const float*)d_in[5];
  const float*     b2    = (const float*)d_in[6];
  const float*     gamma = (const float*)d_in[7];
  const float*     beta  = (const float*)d_in[8];
  const float*     eps   = (const float*)d_in[9];       // [L]
  const float*     linW  = (const float*)d_in[10];
  const float*     linb  = (const float*)d_in[11];

  float* ws    = (float*)d_ws;
  float* agg   = ws;               // [N,H]
  float* hbuf  = ws + 1 * NH;      // [N,H] running node features
  float* t1    = ws + 2 * NH;      // [N,H]
  float* t2    = ws + 3 * NH;      // [N,H]
  float* stats = ws + 4 * NH;      // [256]
  float* sums  = stats + 2 * HH;   // [G,H]
  float* cnts  = sums + GG * HH;   // [G]

  const int threads = 256;
  const int gemm_blocks = NN / 16;                       // 3125, exact
  const int elem_blocks = (NH + threads - 1) / threads;  // 25000
  const int scat_blocks = (EE * 32 + threads - 1) / threads;

  for (int l = 0; l < LL; ++l) {
    const float* hin = (l == 0) ? x : hbuf;
    hipMemsetAsync(agg, 0, (size_t)NH * sizeof(float), stream);
    gin_scatter_add<<<scat_blocks, threads, 0, stream>>>(hin, ei, agg, EE);
    gin_combine<<<elem_blocks, threads, 0, stream>>>(hin, eps, l, agg, NH);
    gin_gemm128_wmma<<<gemm_blocks, threads, 0, stream>>>(
        agg, W1 + (long long)l * HH * HH, b1 + l * HH, t1, NN, /*relu*/1);
    gin_gemm128_wmma<<<gemm_blocks, threads, 0, stream>>>(
        t1, W2 + (long long)l * HH * HH, b2 + l * HH, t2, NN, /*relu*/1);
    hipMemsetAsync(stats, 0, 2 * HH * sizeof(float), stream);
    gin_bn_stats<<<512, HH, 0, stream>>>(t2, stats, NN);
    gin_bn_apply<<<elem_blocks, threads, 0, stream>>>(
        t2, stats, gamma + l * HH, beta + l * HH, hbuf, NN, l > 0 ? 1 : 0);
  }

  // Readout: y = tanh(h @ lin_W + lin_b) -> mean pool per graph
  gin_gemm128_wmma<<<gemm_blocks, threads, 0, stream>>>(
      hbuf, linW, linb, t1, NN, /*tanh*/2);
  hipMemsetAsync(sums, 0, (size_t)(GG * HH + GG) * sizeof(float), stream);
  gin_pool_accum<<<elem_blocks, threads, 0, stream>>>(t1, batch, sums, NN);
  gin_pool_count<<<(NN + threads - 1) / threads, threads, 0, stream>>>(batch, cnts, NN);
  gin_pool_final<<<(GG * HH + threads - 1) / threads, threads, 0, stream>>>(
      sums, cnts, (float*)d_out);
}